// SMPLRendererV2_10350871183723
// MI455X (gfx1250) — compile-verified
//
#include <hip/hip_runtime.h>
#include <hip/hip_bf16.h>

typedef __attribute__((ext_vector_type(2))) float v2f;
typedef __attribute__((ext_vector_type(8))) float v8f;

#define NF        27552
#define NBLK      6888          // NF/4 face-blocks (4 faces per WMMA)
#define NBLK_PAD  6912          // padded to 108 chunks of 64 blocks
#define NCHUNK    108
#define RCH       64            // face-blocks per LDS chunk (16 KB)
#define ISZ       256
#define NEARV     0.1f
#define FARV      100.0f
#define DEN_LO    0.01f         // 1/FAR  : zp<FAR  <=> den>0.01
#define DEN_HI    10.0f         // 1/NEAR : zp>NEAR <=> den<10
#define ZOFF      2.7320508075688772f   // 1 + sqrt(3) = -EYE.z
#define DEADC     (-1.0e30f)

// ---------------------------------------------------------------------------
// Tensor Data Mover staging (gfx1250). Guarded so the file compiles on both
// toolchains; falls back to manual float4 -> LDS copies if absent.
// ---------------------------------------------------------------------------
#if defined(__has_builtin)
#if __has_builtin(__builtin_amdgcn_tensor_load_to_lds)
#define USE_TDM 1
#endif
#endif

#ifdef USE_TDM
#if defined(__has_include)
#if __has_include(<hip/amd_detail/amd_gfx1250_TDM.h>)
#define TDM_6ARG 1      // therock-10.0 headers -> clang-23 6-arg builtin
#endif
#endif
typedef unsigned int v4u __attribute__((ext_vector_type(4)));
typedef int          v4i __attribute__((ext_vector_type(4)));
typedef int          v8i __attribute__((ext_vector_type(8)));

__device__ __forceinline__ unsigned lds_byte_offset(const void* p) {
  // generic -> LDS address space; ptrtoint gives the byte offset in LDS
  return (unsigned)(unsigned long long)(__attribute__((address_space(3))) const char*)p;
}

// 1-D TDM tile: 4096 x 4B contiguous elements (16 KB) global -> LDS
__device__ __forceinline__ void tdm_load_16kb(const float* gsrc, unsigned ldsOff) {
  const unsigned long long ga = (unsigned long long)gsrc;
  v4u g0;
  g0[0] = 1u;                                            // count=1, user desc
  g0[1] = ldsOff;                                        // lds_addr (bytes)
  g0[2] = (unsigned)(ga & 0xffffffffull);                // global_addr[31:0]
  g0[3] = (unsigned)((ga >> 32) & 0x01ffffffull)         // global_addr[56:32]
        | 0x80000000u;                                   // type=2 ("image")
  v8i g1;
  g1[0] = (int)(2u << 16);    // workgroup_mask=0, data_size=2 (4 bytes)
  g1[1] = (int)(4096u << 16); // tensor_dim0[15:0]=4096 in bits[31:16]
  g1[2] = (int)(1u << 16);    // tensor_dim0[31:16]=0 | tensor_dim1[15:0]=1
  g1[3] = (int)(4096u << 16); // tensor_dim1[31:16]=0 | tile_dim0=4096
  g1[4] = 1;                  // tile_dim1=1, tile_dim2=0
  g1[5] = 4096;               // tensor_dim0_stride[31:0]
  g1[6] = 0;                  // stride hi / tensor_dim1_stride lo
  g1[7] = 0;
  v4i z4 = {0, 0, 0, 0};
#ifdef TDM_6ARG
  v8i z8 = {0, 0, 0, 0, 0, 0, 0, 0};
  __builtin_amdgcn_tensor_load_to_lds(g0, g1, z4, z4, z8, 0);
#else
  __builtin_amdgcn_tensor_load_to_lds(g0, g1, z4, z4, 0);
#endif
}
#endif // USE_TDM

// ---------------------------------------------------------------------------
// Kernel 1: per-face preprocessing (unchanged from round 1).
// ---------------------------------------------------------------------------
__global__ __launch_bounds__(256) void smpl_prep(
    const float* __restrict__ cam,
    const float* __restrict__ verts,
    const float* __restrict__ img,
    const int*   __restrict__ faces,
    float* __restrict__ texOut,   // d_out + 3*ISZ*ISZ, NF*81 floats
    float* __restrict__ packA,    // NBLK_PAD*64 floats
    float* __restrict__ fzs)      // NF*3 floats
{
  const int f = blockIdx.x * 256 + threadIdx.x;
  if (f >= NBLK_PAD * 4) return;
  const int blk = f >> 2, jl = f & 3;
  float* pa = packA + blk * 64;

  float rx[4], ry[4], rc[4];

  if (f >= NF) {
    #pragma unroll
    for (int e = 0; e < 4; ++e) { rx[e] = 0.f; ry[e] = 0.f; rc[e] = (e < 3) ? DEADC : 0.f; }
  } else {
    const float sc = cam[0], tx = cam[1], ty = cam[2];
    const int ia = faces[3*f+0], ib = faces[3*f+1], ic = faces[3*f+2];
    const float ax = verts[3*ia], ay = verts[3*ia+1], az = verts[3*ia+2];
    const float bx = verts[3*ib], by = verts[3*ib+1], bz = verts[3*ib+2];
    const float cxv = verts[3*ic], cyv = verts[3*ic+1], czv = verts[3*ic+2];

    const float p0x = sc*(ax+tx),  p0y = sc*(ay+ty);
    const float p1x = sc*(bx+tx),  p1y = sc*(by+ty);
    const float p2x = sc*(cxv+tx), p2y = sc*(cyv+ty);

    // ---- texture extraction: 3x3 samples, bilinear from uv_imgs ----
    {
      const float dx0 = p0x - p2x, dy0 = p0y - p2y;
      const float dx1 = p1x - p2x, dy1 = p1y - p2y;
      const float abv[3] = {0.f, 0.5f, 1.f};
      float* to = texOut + (size_t)f * 81;
      #pragma unroll
      for (int i = 0; i < 3; ++i) {
        #pragma unroll
        for (int j2 = 0; j2 < 3; ++j2) {
          float gx = p2x + abv[i]*dx0 + abv[j2]*dx1;
          float gy = p2y + abv[i]*dy0 + abv[j2]*dy1;
          gx = fminf(fmaxf(gx, -1.f), 1.f);
          gy = fminf(fmaxf(gy, -1.f), 1.f);
          const float x = (gx + 1.f) * 0.5f * 255.f;
          const float y = (gy + 1.f) * 0.5f * 255.f;
          const float xf = floorf(x), yf = floorf(y);
          const float wx = x - xf,   wy = y - yf;
          const int x0 = (int)fminf(fmaxf(xf,       0.f), 255.f);
          const int x1 = (int)fminf(fmaxf(xf + 1.f, 0.f), 255.f);
          const int y0 = (int)fminf(fmaxf(yf,       0.f), 255.f);
          const int y1 = (int)fminf(fmaxf(yf + 1.f, 0.f), 255.f);
          const float w00 = (1.f-wx)*(1.f-wy), w10 = wx*(1.f-wy);
          const float w01 = (1.f-wx)*wy,       w11 = wx*wy;
          #pragma unroll
          for (int c = 0; c < 3; ++c) {
            const float* ip = img + c * (ISZ*ISZ);
            const float v = ip[y0*ISZ + x0]*w00 + ip[y0*ISZ + x1]*w10
                          + ip[y1*ISZ + x0]*w01 + ip[y1*ISZ + x1]*w11;
            float* t0 = to + i*27 + j2*9 + c;
            t0[0] = v; t0[3] = v; t0[6] = v;   // broadcast over 3rd TS axis
          }
        }
      }
    }

    // ---- rasterization coefficients ----
    const float X0 = p0x, Y0 = -p0y, Z0 = az + ZOFF;
    const float X1 = p1x, Y1 = -p1y, Z1 = bz + ZOFF;
    const float X2 = p2x, Y2 = -p2y, Z2 = czv + ZOFF;
    const float a0 = Y1-Y2, a1 = Y2-Y0, a2 = Y0-Y1;
    const float b0 = X2-X1, b1 = X0-X2, b2 = X1-X0;
    const float c0 = X1*Y2 - X2*Y1, c1 = X2*Y0 - X0*Y2, c2 = X0*Y1 - X1*Y0;
    const float det = c0 + c1 + c2;
    const bool  front = det > 0.f;
    const float det_safe = (fabsf(det) > 1e-10f) ? det : 1e-10f;
    const float zs0 = (fabsf(Z0) > 1e-6f) ? Z0 : 1e-6f;
    const float zs1 = (fabsf(Z1) > 1e-6f) ? Z1 : 1e-6f;
    const float zs2 = (fabsf(Z2) > 1e-6f) ? Z2 : 1e-6f;
    fzs[3*f+0] = zs0; fzs[3*f+1] = zs1; fzs[3*f+2] = zs2;

    if (!front) {
      #pragma unroll
      for (int e = 0; e < 4; ++e) { rx[e] = 0.f; ry[e] = 0.f; rc[e] = (e < 3) ? DEADC : 0.f; }
    } else {
      const float inv = 1.f / det_safe;
      const float iz0 = 1.f/zs0, iz1 = 1.f/zs1, iz2 = 1.f/zs2;
      rx[0] = a0*inv; ry[0] = b0*inv; rc[0] = c0*inv;
      rx[1] = a1*inv; ry[1] = b1*inv; rc[1] = c1*inv;
      rx[2] = a2*inv; ry[2] = b2*inv; rc[2] = c2*inv;
      rx[3] = (a0*iz0 + a1*iz1 + a2*iz2) * inv;
      ry[3] = (b0*iz0 + b1*iz1 + b2*iz2) * inv;
      rc[3] = (c0*iz0 + c1*iz1 + c2*iz2) * inv;
    }
  }

  #pragma unroll
  for (int e = 0; e < 4; ++e) {
    const int m = 4*jl + e;
    pa[2*m]          = rx[e];   // K0 (coefX), lanes 0-15
    pa[2*m + 1]      = ry[e];   // K1 (coefY)
    pa[2*(16+m)]     = rc[e];   // K2 (coefC), lanes 16-31
    pa[2*(16+m) + 1] = 0.f;     // K3
  }
}

// ---------------------------------------------------------------------------
// Kernel 2: z-buffer rasterization via V_WMMA_F32_16X16X4_F32.
// Division-free hot loop: argmin(1/den) == argmax(den); NEAR/FAR window
// becomes den in (0.01, 10).  Face data streamed via TDM (or float4 copies)
// into double-buffered 16KB LDS chunks.
// ---------------------------------------------------------------------------
__global__ __launch_bounds__(256) void smpl_raster(
    const float* __restrict__ packA,
    const float* __restrict__ fzs,
    const float* __restrict__ tex,     // textures output region (read back)
    float* __restrict__ images)
{
  __shared__ alignas(16) float lds[2][RCH * 64];   // 2 x 16 KB
  const int tid  = threadIdx.x;
  const int wave = tid >> 5;
  const int lane = tid & 31;
  const int n    = lane & 15;
  const bool hiHalf = lane >= 16;
  const int pix = (blockIdx.x * 8 + wave) * 16 + n;
  const int ix = pix & (ISZ - 1);
  const int iy = pix >> 8;
  const float xp = (2.f*(float)ix + 1.f - (float)ISZ) * (1.f/(float)ISZ);
  const float yp = (2.f*(float)iy + 1.f - (float)ISZ) * (1.f/(float)ISZ);

  // B operand (4x16): lanes 0-15 -> (K0,K1)=(xp,yp); lanes 16-31 -> (K2,K3)=(1,0)
  v2f B;
  B.x = hiHalf ? 1.f : xp;
  B.y = hiHalf ? 0.f : yp;

  float bestDen = 0.f;   // max den over valid faces; 0 => no hit
  int   bestF   = 0;

  // ---- preload chunk 0 ----
#ifdef USE_TDM
  if (wave == 0) {
    tdm_load_16kb(packA, lds_byte_offset(&lds[0][0]));
    __builtin_amdgcn_s_wait_tensorcnt(0);
  }
#else
  {
    const float4* src = (const float4*)packA;
    float4* dst = (float4*)(&lds[0][0]);
    #pragma unroll
    for (int k = 0; k < 4; ++k) dst[tid + k*256] = src[tid + k*256];
  }
#endif
  __syncthreads();

  for (int ch = 0; ch < NCHUNK; ++ch) {
    const int cur = ch & 1;
    // stage next chunk into the other buffer (overlaps compute)
#ifdef USE_TDM
    if (wave == 0 && ch + 1 < NCHUNK) {
      tdm_load_16kb(packA + (size_t)(ch + 1) * RCH * 64,
                    lds_byte_offset(&lds[cur ^ 1][0]));
    }
#else
    if (ch + 1 < NCHUNK) {
      const float4* src = (const float4*)(packA + (size_t)(ch + 1) * RCH * 64);
      float4* dst = (float4*)(&lds[cur ^ 1][0]);
      #pragma unroll
      for (int k = 0; k < 4; ++k) dst[tid + k*256] = src[tid + k*256];
    }
#endif
    if (ch + 2 < NCHUNK) {
      __builtin_prefetch(packA + (size_t)(ch + 2) * RCH * 64 + tid * 16, 0, 0);
    }

    const float* base = &lds[cur][0];
    #pragma unroll 4
    for (int b = 0; b < RCH; ++b) {
      const v2f A = *(const v2f*)(base + (b << 6) + (lane << 1));
      v8f c = {0.f, 0.f, 0.f, 0.f, 0.f, 0.f, 0.f, 0.f};
      v8f d = __builtin_amdgcn_wmma_f32_16x16x4_f32(
          false, A, false, B, (short)0, c, false, false);
      const int fbase = ((ch * RCH + b) << 2) + (hiHalf ? 2 : 0);
      #pragma unroll
      for (int jj = 0; jj < 2; ++jj) {
        const float w0 = d[jj*4+0], w1 = d[jj*4+1], w2 = d[jj*4+2], den = d[jj*4+3];
        const float mn = fminf(w0, fminf(w1, w2));
        const float mx = fmaxf(w0, fmaxf(w1, w2));
        const bool ok = (mn >= 0.f) && (mx <= 1.f) && (den > DEN_LO) && (den < DEN_HI);
        const float cand = ok ? den : 0.f;
        if (cand > bestDen) { bestDen = cand; bestF = fbase + jj; }
      }
    }
#ifdef USE_TDM
    if (wave == 0) __builtin_amdgcn_s_wait_tensorcnt(0);
#endif
    __syncthreads();
  }

  // merge the two lane-halves (faces {0,1} vs {2,3} mod 4 of the same pixel)
  {
    const float oD = __shfl_xor(bestDen, 16, 32);
    const int   oF = __shfl_xor(bestF, 16, 32);
    if (oD > bestDen || (oD == bestDen && oF < bestF)) { bestDen = oD; bestF = oF; }
  }

  if (lane < 16) {
    float col0 = 0.f, col1 = 0.f, col2 = 0.f;
    if (bestDen > 0.f) {
      const int f = bestF;
      const int blk = f >> 2, jl = f & 3;
      const float* pa = packA + blk * 64;
      float wc[3]; float s = 0.f;
      #pragma unroll
      for (int e = 0; e < 3; ++e) {
        const int m = 4*jl + e;
        const float w = pa[2*m]*xp + pa[2*m+1]*yp + pa[2*(16+m)];
        const float v = w / fzs[3*f + e];
        wc[e] = v; s += v;
      }
      s = fmaxf(s, 1e-8f);
      int lo[3], hi2[3]; float fr[3];
      #pragma unroll
      for (int e = 0; e < 3; ++e) {
        float t = wc[e] / s;
        t = fminf(fmaxf(t, 0.f), 1.f) * 2.f;
        const float lf = fminf(fmaxf(floorf(t), 0.f), 2.f);
        fr[e] = t - lf;
        lo[e] = (int)lf;
        hi2[e] = (lo[e] + 1 < 2) ? lo[e] + 1 : 2;
      }
      // textures are constant along the 3rd TS axis -> its weights sum to 1
      const float* tf = tex + (size_t)f * 81;
      #pragma unroll
      for (int b0 = 0; b0 < 2; ++b0) {
        const int i0 = b0 ? hi2[0] : lo[0];
        const float g0 = b0 ? fr[0] : (1.f - fr[0]);
        #pragma unroll
        for (int b1 = 0; b1 < 2; ++b1) {
          const int i1 = b1 ? hi2[1] : lo[1];
          const float g = g0 * (b1 ? fr[1] : (1.f - fr[1]));
          const float* tp = tf + i0*27 + i1*9;
          col0 += g * tp[0]; col1 += g * tp[1]; col2 += g * tp[2];
        }
      }
    }
    const int po = iy * ISZ + ix;
    images[po]             = col0;
    images[ISZ*ISZ + po]   = col1;
    images[2*ISZ*ISZ + po] = col2;
  }
}

// ---------------------------------------------------------------------------
extern "C" void kernel_launch(void* const* d_in, const int* in_sizes, int n_in,
                              void* d_out, int out_size, void* d_ws, size_t ws_size,
                              hipStream_t stream) {
  const float* cam   = (const float*)d_in[0];   // (1,3)
  const float* verts = (const float*)d_in[1];   // (1,6890,3)
  const float* img   = (const float*)d_in[2];   // (1,3,256,256)
  const int*   faces = (const int*)d_in[3];     // (27552,3)

  float* images = (float*)d_out;                 // (1,3,256,256)
  float* texOut = images + 3 * ISZ * ISZ;        // (1,27552,3,3,3,3)

  float* packA = (float*)d_ws;                   // NBLK_PAD*64 floats
  float* fzs   = packA + (size_t)NBLK_PAD * 64;  // NF*3 floats

  smpl_prep<<<(NBLK_PAD * 4) / 256, 256, 0, stream>>>(
      cam, verts, img, faces, texOut, packA, fzs);
  smpl_raster<<<(ISZ * ISZ) / 128, 256, 0, stream>>>(
      packA, fzs, texOut, images);
}